// KPConvSimpleBlock_second_76227079570100
// MI455X (gfx1250) — compile-verified
//
#include <hip/hip_runtime.h>

// KPConv fused kernel for MI455X (gfx1250, wave32).
// Pipeline: prep_w -> kpconv_fused (WMMA f32 16x16x4) -> bn_stats -> bn_apply.

#define K_POINTS 15
#define IN_CH    6
#define OUT_CH   128
#define S_NB     32
#define KPAD     48          // reduced contraction dim: 15 kpoints * 3 xyz ch = 45, padded to 48
#define MTILE    16
#define QTILES   4           // query tiles per workgroup
#define NQ       16384       // b * ng
#define NPTS     65536       // b * n
#define NEG_SLOPE 0.2f
#define BN_EPS    1e-5f

typedef __attribute__((ext_vector_type(2))) float v2f;
typedef __attribute__((ext_vector_type(8))) float v8f;

// ---------------------------------------------------------------------------
// Pack W[15][6][128] -> Wp[48][128]: keep only channels 3..5 (channels 0..2
// multiply identically-zero features), pad 45 -> 48 rows with zeros.
// ---------------------------------------------------------------------------
__global__ __launch_bounds__(256) void prep_w(const float* __restrict__ W,
                                              float* __restrict__ Wp) {
    int i = blockIdx.x * 256 + threadIdx.x;          // 48*128 = 6144
    if (i >= KPAD * OUT_CH) return;
    int row = i >> 7;
    int o   = i & (OUT_CH - 1);
    float v = 0.0f;
    if (row < K_POINTS * 3) {
        int k = row / 3, c = row - 3 * k;
        v = W[(k * IN_CH + (c + 3)) * OUT_CH + o];
    }
    Wp[row * OUT_CH + o] = v;
}

// ---------------------------------------------------------------------------
// Fused gather + kernel-point influence + weighted aggregation + GEMM.
// Block: 256 threads (8 wave32). Each block handles QTILES tiles of 16
// queries; each wave owns one 16-wide output-channel tile (8*16 = 128 = OUT_CH).
// Writes raw (pre-BN) output to outraw.
// ---------------------------------------------------------------------------
__global__ __launch_bounds__(256) void kpconv_fused(
    const float* __restrict__ xyz,      // [NPTS,3]
    const float* __restrict__ centors,  // [NQ,3]
    const int*   __restrict__ idx,      // [NQ*S_NB]
    const float* __restrict__ Kp,       // [15,3]
    const float* __restrict__ Wp,       // [48,128] packed+padded
    float*       __restrict__ outraw)   // [NQ,128]
{
    __shared__ float Wlds[KPAD][OUT_CH];         // 24 KB, resident all tiles
    __shared__ float nbl[MTILE][S_NB][3];        // 6 KB, centered neighbors
    __shared__ float qx[MTILE][3];               // query centers
    __shared__ float Alds[MTILE][KPAD + 1];      // A-tile, +1 pad vs 64 banks
    __shared__ float Kpl[K_POINTS * 3];          // kernel point dispositions

    const int tid  = threadIdx.x;
    const int lane = tid & 31;
    const int wave = tid >> 5;                   // 0..7 -> output N-tile

    // Stage 0: stage packed weights + kernel points into LDS once.
    for (int i = tid; i < KPAD * OUT_CH; i += 256)
        Wlds[i >> 7][i & (OUT_CH - 1)] = Wp[i];
    if (tid < K_POINTS * 3)
        Kpl[tid] = Kp[tid];
    __syncthreads();

    const int qbase0 = blockIdx.x * (MTILE * QTILES);

    for (int t = 0; t < QTILES; ++t) {
        const int qbase = qbase0 + t * MTILE;

        // Prefetch next tile's neighbor indices (global_prefetch_b8).
        if (t + 1 < QTILES)
            __builtin_prefetch(&idx[(qbase + MTILE) * S_NB], 0, 0);

        // Stage 1a: load query centers.
        if (tid < MTILE * 3)
            (&qx[0][0])[tid] = centors[qbase * 3 + tid];
        __syncthreads();

        // Stage 1b: gather neighbor points, center on query (512 pairs).
        for (int p = tid; p < MTILE * S_NB; p += 256) {
            int row = p >> 5;
            int snb = p & (S_NB - 1);
            int gi  = idx[(qbase + row) * S_NB + snb];
            float px, py, pz;
            if (gi >= 0 && gi < NPTS) {
                px = xyz[gi * 3 + 0];
                py = xyz[gi * 3 + 1];
                pz = xyz[gi * 3 + 2];
            } else {                              // shadow neighbor
                px = py = pz = 1e6f;
            }
            nbl[row][snb][0] = px - qx[row][0];
            nbl[row][snb][1] = py - qx[row][1];
            nbl[row][snb][2] = pz - qx[row][2];
        }
        __syncthreads();

        // Stage 2: per (query-row, kernel-point) linear-influence aggregation.
        // 16 rows x 16 slots (15 kpoints + 1 pad slot) = 256 cells = 1/thread.
        {
            int row = tid >> 4;
            int k   = tid & 15;
            if (k < K_POINTS) {
                float kx = Kpl[k * 3 + 0], ky = Kpl[k * 3 + 1], kz = Kpl[k * 3 + 2];
                float q0 = qx[row][0], q1 = qx[row][1], q2 = qx[row][2];
                float ax = 0.0f, ay = 0.0f, az = 0.0f;
                #pragma unroll 4
                for (int s = 0; s < S_NB; ++s) {
                    float nx = nbl[row][s][0];
                    float ny = nbl[row][s][1];
                    float nz = nbl[row][s][2];
                    float dx = nx - kx, dy = ny - ky, dz = nz - kz;
                    float d  = sqrtf(fmaf(dx, dx, fmaf(dy, dy, dz * dz)));
                    float w  = fmaxf(1.0f - d, 0.0f);      // POINT_INFLUENCE = 1
                    // feature channels 3..5 are the *uncentered* coords
                    ax = fmaf(w, nx + q0, ax);
                    ay = fmaf(w, ny + q1, ay);
                    az = fmaf(w, nz + q2, az);
                }
                Alds[row][k * 3 + 0] = ax;
                Alds[row][k * 3 + 1] = ay;
                Alds[row][k * 3 + 2] = az;
            } else {                               // zero the pad columns 45..47
                Alds[row][45] = 0.0f;
                Alds[row][46] = 0.0f;
                Alds[row][47] = 0.0f;
            }
        }
        __syncthreads();

        // Stage 3: 16x128 GEMM tile via v_wmma_f32_16x16x4_f32.
        // A: 16x48 in Alds; B: 48x128 in Wlds; wave 'wave' owns cols
        // [wave*16, wave*16+16). K-loop: 12 steps of 4.
        {
            v8f acc = {};
            const int mrow  = lane & 15;
            const int khalf = (lane >> 4) << 1;            // 0 or 2
            const int ncol  = wave * 16 + (lane & 15);
            #pragma unroll
            for (int kk = 0; kk < KPAD / 4; ++kk) {
                const int kb = kk * 4 + khalf;
                v2f a, b;
                a.x = Alds[mrow][kb + 0];
                a.y = Alds[mrow][kb + 1];
                b.x = Wlds[kb + 0][ncol];
                b.y = Wlds[kb + 1][ncol];
                acc = __builtin_amdgcn_wmma_f32_16x16x4_f32(
                    /*neg_a=*/false, a, /*neg_b=*/false, b,
                    /*c_mod=*/(short)0, acc,
                    /*reuse_a=*/false, /*reuse_b=*/false);
            }
            // C/D layout: VGPR r -> M = r (lanes 0-15) or 8+r (lanes 16-31),
            // N = lane & 15 within the tile.
            const int mhi = (lane >> 4) << 3;              // 0 or 8
            #pragma unroll
            for (int r = 0; r < 8; ++r) {
                int m = qbase + mhi + r;
                outraw[m * OUT_CH + ncol] = acc[r];
            }
        }
        __syncthreads();
    }
}

// ---------------------------------------------------------------------------
// BatchNorm statistics: one block per channel, deterministic tree reduction.
// Emits per-channel scale = gamma * rsqrt(var + eps), shift = beta - mean*scale.
// ---------------------------------------------------------------------------
__global__ __launch_bounds__(256) void bn_stats(const float* __restrict__ outraw,
                                                const float* __restrict__ gamma,
                                                const float* __restrict__ beta,
                                                float* __restrict__ scale,
                                                float* __restrict__ shift) {
    __shared__ float ssum[256];
    __shared__ float ssq[256];
    const int ch = blockIdx.x;
    float sum = 0.0f, sq = 0.0f;
    for (int r = threadIdx.x; r < NQ; r += 256) {
        float v = outraw[r * OUT_CH + ch];
        sum += v;
        sq = fmaf(v, v, sq);
    }
    ssum[threadIdx.x] = sum;
    ssq[threadIdx.x] = sq;
    __syncthreads();
    for (int off = 128; off > 0; off >>= 1) {
        if (threadIdx.x < off) {
            ssum[threadIdx.x] += ssum[threadIdx.x + off];
            ssq[threadIdx.x]  += ssq[threadIdx.x + off];
        }
        __syncthreads();
    }
    if (threadIdx.x == 0) {
        float inv_n = 1.0f / (float)NQ;
        float mean  = ssum[0] * inv_n;
        float var   = fmaf(-mean, mean, ssq[0] * inv_n);   // biased variance
        float sc    = gamma[ch] * rsqrtf(var + BN_EPS);
        scale[ch]   = sc;
        shift[ch]   = fmaf(-mean, sc, beta[ch]);
    }
}

// ---------------------------------------------------------------------------
// In-place affine BN + LeakyReLU on d_out.
// ---------------------------------------------------------------------------
__global__ __launch_bounds__(256) void bn_apply(float* __restrict__ out,
                                                const float* __restrict__ scale,
                                                const float* __restrict__ shift) {
    int base = (blockIdx.x * 256 + threadIdx.x) * 4;
    #pragma unroll
    for (int j = 0; j < 4; ++j) {
        int e = base + j;
        int c = e & (OUT_CH - 1);
        float v = fmaf(out[e], scale[c], shift[c]);
        out[e] = (v >= 0.0f) ? v : NEG_SLOPE * v;
    }
}

// ---------------------------------------------------------------------------
extern "C" void kernel_launch(void* const* d_in, const int* in_sizes, int n_in,
                              void* d_out, int out_size, void* d_ws, size_t ws_size,
                              hipStream_t stream) {
    const float* xyz     = (const float*)d_in[0];   // [4,16384,3]
    const float* centors = (const float*)d_in[1];   // [4,4096,3]
    const int*   idx     = (const int*)d_in[2];     // [4*4096*32]
    const float* Kp      = (const float*)d_in[3];   // [15,3]
    const float* W       = (const float*)d_in[4];   // [15,6,128]
    const float* gamma   = (const float*)d_in[5];   // [128]
    const float* beta    = (const float*)d_in[6];   // [128]
    float* out = (float*)d_out;                     // [4,4096,128] (raw then final)

    float* Wp    = (float*)d_ws;                    // 48*128 floats packed W
    float* scale = Wp + KPAD * OUT_CH;              // 128
    float* shift = scale + OUT_CH;                  // 128

    prep_w<<<(KPAD * OUT_CH + 255) / 256, 256, 0, stream>>>(W, Wp);
    kpconv_fused<<<NQ / (MTILE * QTILES), 256, 0, stream>>>(
        xyz, centors, idx, Kp, Wp, out);
    bn_stats<<<OUT_CH, 256, 0, stream>>>(out, gamma, beta, scale, shift);
    bn_apply<<<(NQ * OUT_CH) / (256 * 4), 256, 0, stream>>>(out, scale, shift);
}